// ECELoss_22969485099012
// MI455X (gfx1250) — compile-verified
//
#include <hip/hip_runtime.h>
#include <math.h>

#define NBINS 15
#define NCLS  100

typedef __attribute__((ext_vector_type(4))) float f4;

// ---------------------------------------------------------------- zero bins
__global__ void ece_zero(float* __restrict__ bins) {
    if (threadIdx.x < 3 * NBINS) bins[threadIdx.x] = 0.0f;
}

// ---------------------------------------------------------------- main pass
// One wave32 per row: 25 lanes issue one nontemporal global_load_b128 each
// (400B/row, coalesced, 16B aligned). Register-level software pipeline holds
// the next row; global_prefetch_b8 runs one further stage ahead (row+2*stride)
// to warm L2 while the in-flight load completes. Wave32 5-step butterflies
// for (max, first-index) and sum-of-exp. Per-block LDS histogram, one
// 45-float global atomic flush per block.
__global__ void __launch_bounds__(256)
ece_main(const float* __restrict__ logits,
         const int*   __restrict__ labels,
         float*       __restrict__ bins,
         int n) {
    __shared__ float sb[3 * NBINS];
    for (int i = threadIdx.x; i < 3 * NBINS; i += blockDim.x) sb[i] = 0.0f;
    __syncthreads();

    const int lane   = threadIdx.x & 31;
    const int wave   = (blockIdx.x * blockDim.x + threadIdx.x) >> 5;
    const int nwaves = (gridDim.x * blockDim.x) >> 5;
    const int col    = lane * 4;                 // lanes 0..24 cover 100 cols
    const bool active = (col < NCLS);

    const f4 NEGINF = (f4){-INFINITY, -INFINITY, -INFINITY, -INFINITY};

    // Per-wave base pointer for this lane's slice; advance by a fixed byte
    // stride each iteration so loop address math is one 64-bit add.
    const size_t stride_elts = (size_t)nwaves * NCLS;
    const float* lp = logits + (size_t)wave * NCLS + col;

    int row = wave;
    f4 v = NEGINF;
    if (row < n && active) v = __builtin_nontemporal_load((const f4*)lp);

    while (row < n) {
        // ---- software pipeline: issue next row's load before reducing ----
        const int nrow = row + nwaves;
        const float* nlp = lp + stride_elts;
        f4 nv = NEGINF;
        if (nrow < n && active) {
            nv = __builtin_nontemporal_load((const f4*)nlp);
            // warm L2 one stage beyond the register pipeline
            if (nrow + nwaves < n)
                __builtin_prefetch(nlp + stride_elts, 0, 0); // global_prefetch_b8
        }

        // ---- lane-local max + first-index argmax over 4 elements ----
        float m = v.x; int a = col;
        if (v.y > m) { m = v.y; a = col + 1; }
        if (v.z > m) { m = v.z; a = col + 2; }
        if (v.w > m) { m = v.w; a = col + 3; }

        // ---- wave32 butterfly: (max, lowest index on tie) ----
        #pragma unroll
        for (int off = 16; off >= 1; off >>= 1) {
            float om = __shfl_xor(m, off, 32);
            int   oa = __shfl_xor(a, off, 32);
            if (om > m || (om == m && oa < a)) { m = om; a = oa; }
        }

        // ---- sum of exp(l - m); inactive lanes hold -inf -> exp = 0 ----
        float s = __expf(v.x - m) + __expf(v.y - m)
                + __expf(v.z - m) + __expf(v.w - m);
        #pragma unroll
        for (int off = 16; off >= 1; off >>= 1)
            s += __shfl_xor(s, off, 32);

        if (lane == 0) {
            const float conf = 1.0f / s;               // exp(m)/sum(exp) = 1/s
            int bin = (int)ceilf(conf * (float)NBINS) - 1;
            bin = bin < 0 ? 0 : (bin > NBINS - 1 ? NBINS - 1 : bin);
            const float acc = (labels[row] == a) ? 1.0f : 0.0f;
            atomicAdd(&sb[bin],             1.0f);
            atomicAdd(&sb[NBINS + bin],     conf);
            atomicAdd(&sb[2 * NBINS + bin], acc);
        }

        v   = nv;
        row = nrow;
        lp  = nlp;
    }

    __syncthreads();
    for (int i = threadIdx.x; i < 3 * NBINS; i += blockDim.x) {
        const float partial = sb[i];
        if (partial != 0.0f) atomicAdd(&bins[i], partial);
    }
}

// ---------------------------------------------------------------- finalize
__global__ void ece_final(const float* __restrict__ bins,
                          float* __restrict__ out, float inv_n) {
    if (threadIdx.x == 0) {
        float ece = 0.0f;
        #pragma unroll
        for (int i = 0; i < NBINS; ++i) {
            const float c = bins[i];
            if (c > 0.0f) {
                const float cs = bins[NBINS + i];
                const float as = bins[2 * NBINS + i];
                ece += fabsf(cs / c - as / c) * (c * inv_n);
            }
        }
        out[0] = ece;
    }
}

// ---------------------------------------------------------------- launcher
extern "C" void kernel_launch(void* const* d_in, const int* in_sizes, int n_in,
                              void* d_out, int out_size, void* d_ws, size_t ws_size,
                              hipStream_t stream) {
    const float* logits = (const float*)d_in[0];
    const int*   labels = (const int*)d_in[1];
    float*       outp   = (float*)d_out;
    float*       bins   = (float*)d_ws;     // 45 floats of scratch

    const int n = in_sizes[1];              // number of rows (labels count)

    ece_zero<<<1, 64, 0, stream>>>(bins);

    const int blocks = 2048;                // 16384 waves, ~128 rows each
    ece_main<<<blocks, 256, 0, stream>>>(logits, labels, bins, n);

    ece_final<<<1, 32, 0, stream>>>(bins, outp, 1.0f / (float)n);
}